// MyChamferDistance_11493332484300
// MI455X (gfx1250) — compile-verified
//
#include <hip/hip_runtime.h>
#include <math.h>

typedef __attribute__((ext_vector_type(2))) float v2f;
typedef __attribute__((ext_vector_type(8))) float v8f;

__global__ void zero_out_kernel(float* out) {
    if (threadIdx.x == 0 && blockIdx.x == 0) out[0] = 0.0f;
}

// One wave (32 lanes) per (batch, 16-row tile of query cloud Q).
// For each of the 16 query points, computes min over all NR database points R
// of squared distance using V_WMMA_F32_16X16X4_F32 for the -2*q.r term,
// then sqrt once per row, scales and atomically accumulates into *out.
__global__ __launch_bounds__(256) void chamfer_pass_kernel(
    const float* __restrict__ Q,   // [B, NQ, 3]
    const float* __restrict__ R,   // [B, NR, 3]
    float* __restrict__ out,       // scalar accumulator
    int B, int NQ, int NR, float scale)
{
    const int lane          = threadIdx.x & 31;
    const int waveInBlock   = threadIdx.x >> 5;
    const int wavesPerBlock = blockDim.x >> 5;
    const int gwave = blockIdx.x * wavesPerBlock + waveInBlock;

    const int tilesPerBatch = (NQ + 15) >> 4;
    const int b    = gwave / tilesPerBatch;   // wave-uniform
    const int tile = gwave % tilesPerBatch;   // wave-uniform
    if (b >= B) return;                       // wave-uniform exit (EXEC stays all-1 for WMMA)

    const int h = lane >> 4;    // half of wave: selects K=0/1 vs K=2/3 slice
    const int l = lane & 15;    // row (for A) / column (for B) within tile

    // ---- Load this wave's 16 query points (both halves load row l) ----
    int qrowIdx = tile * 16 + l;
    int qrowClamped = qrowIdx < NQ ? qrowIdx : NQ - 1;
    const size_t qbase = ((size_t)b * NQ + (size_t)qrowClamped) * 3;
    float qx = Q[qbase + 0], qy = Q[qbase + 1], qz = Q[qbase + 2];
    float qq = qx * qx + qy * qy + qz * qz;

    // A-operand, 16x4 f32 (2 VGPRs): lanes 0-15 -> K=0,1 ; lanes 16-31 -> K=2,3(=pad 0)
    v2f Aop;
    Aop.x = h ? qz : qx;
    Aop.y = h ? 0.0f : qy;

    // Each lane needs |q|^2 of the 8 rows it accumulates: rows v + 8*h.
    float qrow2[8];
    #pragma unroll
    for (int v = 0; v < 8; ++v)
        qrow2[v] = __shfl(qq, v + 8 * h, 32);

    float qmin[8];   // running min of (rr - 2*q.r) per owned row
    #pragma unroll
    for (int v = 0; v < 8; ++v) qmin[v] = 3.0e38f;

    const float* Rb = R + (size_t)b * NR * 3;
    const int ntiles = (NR + 15) >> 4;

    #pragma unroll 2
    for (int t = 0; t < ntiles; ++t) {
        int m = t * 16 + l;
        int mc = m < NR ? m : NR - 1;
        const size_t rbase = (size_t)mc * 3;
        float rx = Rb[rbase + 0], ry = Rb[rbase + 1], rz = Rb[rbase + 2];
        float rr = rx * rx + ry * ry + rz * rz;
        if (m >= NR) rr = 1.0e37f;          // poison padded columns out of the min

        // Prefetch a future database tile into cache (global_prefetch_b8).
        if (t + 8 < ntiles)
            __builtin_prefetch((const void*)(Rb + ((size_t)(t + 8) * 16 + l) * 3), 0, 0);

        // B-operand, 4x16 f32 (2 VGPRs): symmetric layout to A.
        v2f Bop;
        Bop.x = h ? rz : rx;
        Bop.y = h ? 0.0f : ry;

        v8f acc = {};
        // D[row=v+8h][col=l] = q[row] . r[col]   (K=4, last K is zero pad)
        acc = __builtin_amdgcn_wmma_f32_16x16x4_f32(
            /*neg_a=*/false, Aop, /*neg_b=*/false, Bop,
            /*c_mod=*/(short)0, acc, /*reuse_a=*/false, /*reuse_b=*/false);

        #pragma unroll
        for (int v = 0; v < 8; ++v) {
            float q2 = fmaf(-2.0f, acc[v], rr);   // rr - 2*dot  (|q|^2 added at the end)
            qmin[v] = fminf(qmin[v], q2);
        }
    }

    // Min-reduce across the 16 lanes of each half (masks < 16 keep halves separate).
    #pragma unroll
    for (int off = 1; off < 16; off <<= 1) {
        #pragma unroll
        for (int v = 0; v < 8; ++v)
            qmin[v] = fminf(qmin[v], __shfl_xor(qmin[v], off, 32));
    }

    // Lane l==0 of each half now holds full row-mins for rows 8h..8h+7.
    if (l == 0) {
        int baseRow = tile * 16 + 8 * h;
        float s = 0.0f;
        #pragma unroll
        for (int v = 0; v < 8; ++v) {
            if (baseRow + v < NQ) {
                float d2 = qrow2[v] + qmin[v];
                s += sqrtf(fmaxf(d2, 0.0f));
            }
        }
        atomicAdd(out, s * scale);
    }
}

extern "C" void kernel_launch(void* const* d_in, const int* in_sizes, int n_in,
                              void* d_out, int out_size, void* d_ws, size_t ws_size,
                              hipStream_t stream) {
    const float* inp = (const float*)d_in[0];   // [B, N, 3] float32
    const float* tgt = (const float*)d_in[1];   // [B, M, 3] float32
    float* out = (float*)d_out;                  // scalar

    const int B = 4, D = 3;                      // per reference setup
    int N = in_sizes[0] / (B * D);
    int M = in_sizes[1] / (B * D);

    zero_out_kernel<<<1, 1, 0, stream>>>(out);

    const int WPB = 8;                  // waves per block
    const int TPB = WPB * 32;           // 256 threads
    int tilesN = (N + 15) / 16;
    int tilesM = (M + 15) / 16;
    int blocks1 = (B * tilesN + WPB - 1) / WPB;
    int blocks2 = (B * tilesM + WPB - 1) / WPB;

    // Direction 1: for each input point, NN distance into target; mean over N.
    chamfer_pass_kernel<<<blocks1, TPB, 0, stream>>>(inp, tgt, out, B, N, M, 1.0f / (float)N);
    // Direction 2: for each target point, NN distance into input; mean over M.
    chamfer_pass_kernel<<<blocks2, TPB, 0, stream>>>(tgt, inp, out, B, M, N, 1.0f / (float)M);
}